// CoVQMF3_51496657879375
// MI455X (gfx1250) — compile-verified
//
#include <hip/hip_runtime.h>

typedef __attribute__((ext_vector_type(2))) float v2f;
typedef __attribute__((ext_vector_type(8))) float v8f;

#define FEAT_C   64
#define FEAT_H   64
#define FEAT_W   64
#define NVX      128   // X voxels
#define NVY      128   // Y voxels
#define NVZ      8     // Z voxels

// ---------------------------------------------------------------------------
// Setup: per (b,n) camera, build 12-float transform block in workspace:
//   rows [0..3]  = pix_T_cam0 row 0  (u = row . [x,y,z,1])
//   rows [4..7]  = pix_T_cam0 row 1  (v)
//   rows [8..11] = camX_T_cam0 row 2 (z; equals pix row 2 since featpix r2=[0,0,1,0])
// safe_inverse: R^T, -R^T t. scale_intrinsics with sx=sy=64/512.
// ---------------------------------------------------------------------------
__global__ void bev_setup_kernel(const float* __restrict__ intrins,
                                 const float* __restrict__ extrins,
                                 float* __restrict__ M) {
    int bn = threadIdx.x;
    if (bn >= 8) return;
    const float* E = extrins + bn * 16;
    const float* K = intrins + bn * 16;
    float cam[3][4];
    #pragma unroll
    for (int i = 0; i < 3; ++i) {
        // rT[i][j] = E[j][i]; t[i] = -sum_j rT[i][j]*E[j][3]
        float t = -(E[0*4+i]*E[0*4+3] + E[1*4+i]*E[1*4+3] + E[2*4+i]*E[2*4+3]);
        cam[i][0] = E[0*4+i];
        cam[i][1] = E[1*4+i];
        cam[i][2] = E[2*4+i];
        cam[i][3] = t;
    }
    const float s = 64.0f / 512.0f;
    float fx = K[0*4+0] * s, fy = K[1*4+1] * s;
    float x0 = K[0*4+2] * s, y0 = K[1*4+2] * s;
    float* o = M + bn * 12;
    #pragma unroll
    for (int k = 0; k < 4; ++k) {
        o[k]     = fx * cam[0][k] + x0 * cam[2][k];  // pix row 0
        o[4 + k] = fy * cam[1][k] + y0 * cam[2][k];  // pix row 1
        o[8 + k] = cam[2][k];                        // camX row 2 (== pix row 2)
    }
}

// ---------------------------------------------------------------------------
// Main: block = (x, z, b); 128 threads = y. WMMA f32 16x16x4 computes the
// 12 projected values (u,v,z for 4 cams) for 16 voxels per issue.
// ---------------------------------------------------------------------------
__global__ __launch_bounds__(128)
void bev_main_kernel(const float* __restrict__ feats,  // (8, 64, 64, 64)
                     const float* __restrict__ M,      // (8, 12) transforms
                     float* __restrict__ out) {        // (2, 64*8, 128, 128)
    const int x    = blockIdx.x;
    const int z    = blockIdx.y;
    const int b    = blockIdx.z;
    const int ty   = threadIdx.x;        // y voxel 0..127
    const int wave = ty >> 5;
    const int lane = ty & 31;
    const int half = (lane < 16) ? 0 : 1;   // which K-half / M-half
    const int n16  = lane & 15;

    const float xc = x * 0.8f - 50.8f;       // X0 + vs/2 = -51.2 + 0.4
    const float zc = z * 0.5f - 2.75f;       // Z0 + vs/2 = -3.0 + 0.25

    __shared__ float trans[4][32][12];       // [wave][voxel-in-wave][u,v,z x 4cams]

    // --- B matrix 4x16: column n<12 -> transform row (cam=n/3, j=n%3).
    // Layout: VGPR0 = K={0|2}, VGPR1 = K={1|3} for lanes {0-15|16-31}.
    v2f Bm;
    {
        int nn  = (n16 < 12) ? n16 : 0;
        int camI = nn / 3, j = nn % 3;
        const float* row = M + (b * 4 + camI) * 12 + j * 4;
        float b0 = row[half ? 2 : 0];
        float b1 = row[half ? 3 : 1];
        if (n16 >= 12) { b0 = 0.0f; b1 = 0.0f; }
        Bm.x = b0; Bm.y = b1;
    }

    // --- Two WMMAs cover the wave's 32 voxels (16 per group).
    const int ywave = ty & ~31;
    #pragma unroll
    for (int g = 0; g < 2; ++g) {
        // A 16x4 row m = (xc, yc_m, zc, 1); lanes0-15 hold K=0,1; lanes16-31 K=2,3
        float yc_m = (float)(ywave + g * 16 + n16) * 0.8f - 50.8f;
        v2f Am;
        Am.x = half ? zc   : xc;
        Am.y = half ? 1.0f : yc_m;
        v8f d = {};
        d = __builtin_amdgcn_wmma_f32_16x16x4_f32(false, Am, false, Bm,
                                                  (short)0, d, false, false);
        // D: lane holds column N=n16, rows M = r + 8*half. Transpose via LDS.
        if (n16 < 12) {
            #pragma unroll
            for (int r = 0; r < 8; ++r)
                trans[wave][g * 16 + half * 8 + r][n16] = d[r];
        }
    }
    __syncthreads();

    // --- Per-camera sampling parameters for this thread's voxel.
    int   offs[4][4];
    float w[4][4];
    #pragma unroll
    for (int c = 0; c < 4; ++c) {
        float u  = trans[wave][lane][c * 3 + 0];
        float v  = trans[wave][lane][c * 3 + 1];
        float zz = trans[wave][lane][c * 3 + 2];
        float den = fmaxf(zz, 1e-6f);
        float px = u / den, py = v / den;
        bool valid = (px > -0.5f) & (px < 63.5f) & (py > -0.5f) & (py < 63.5f)
                   & (zz > 0.0f);
        float sxp = px - 0.5f, syp = py - 0.5f;
        float x0f = floorf(sxp), y0f = floorf(syp);
        float wx = sxp - x0f, wy = syp - y0f;
        int ix0 = (int)fminf(fmaxf(x0f, -8.0f), 72.0f);
        int iy0 = (int)fminf(fmaxf(y0f, -8.0f), 72.0f);
        bool vx0 = (ix0 >= 0) & (ix0 < FEAT_W);
        bool vx1 = (ix0 + 1 >= 0) & (ix0 + 1 < FEAT_W);
        bool vy0 = (iy0 >= 0) & (iy0 < FEAT_H);
        bool vy1 = (iy0 + 1 >= 0) & (iy0 + 1 < FEAT_H);
        int cx0 = min(max(ix0, 0), FEAT_W - 1), cx1 = min(max(ix0 + 1, 0), FEAT_W - 1);
        int cy0 = min(max(iy0, 0), FEAT_H - 1), cy1 = min(max(iy0 + 1, 0), FEAT_H - 1);
        float vf = valid ? 1.0f : 0.0f;          // overall validity folded into weights
        w[c][0] = (1.0f - wx) * (1.0f - wy) * ((vx0 & vy0) ? vf : 0.0f);
        w[c][1] = wx * (1.0f - wy)          * ((vx1 & vy0) ? vf : 0.0f);
        w[c][2] = (1.0f - wx) * wy          * ((vx0 & vy1) ? vf : 0.0f);
        w[c][3] = wx * wy                   * ((vx1 & vy1) ? vf : 0.0f);
        int base = (b * 4 + c) * (FEAT_C * FEAT_H * FEAT_W);  // channel-0 plane
        offs[c][0] = base + cy0 * FEAT_W + cx0;
        offs[c][1] = base + cy0 * FEAT_W + cx1;
        offs[c][2] = base + cy1 * FEAT_W + cx0;
        offs[c][3] = base + cy1 * FEAT_W + cx1;
    }

    // --- Channel loop: gather 4 corners per camera, masked mean over cameras.
    // out[b, ch, z, x, y] flat: ((b*64+ch)*8 + z)*16384 + x*128 + y
    const size_t obase  = ((size_t)(b * 64) * 8 + (size_t)z) * (128 * 128)
                        + (size_t)x * 128 + (size_t)ty;
    #pragma unroll 4
    for (int ch = 0; ch < FEAT_C; ++ch) {
        const float* f = feats + ch * (FEAT_H * FEAT_W);
        float sum = 0.0f, cnt = 0.0f;
        #pragma unroll
        for (int c = 0; c < 4; ++c) {
            float val = w[c][0] * f[offs[c][0]] + w[c][1] * f[offs[c][1]]
                      + w[c][2] * f[offs[c][2]] + w[c][3] * f[offs[c][3]];
            sum += val;
            cnt += (val != 0.0f) ? 1.0f : 0.0f;   // mask = |val| > 0
        }
        out[obase + (size_t)ch * (NVZ * 128 * 128)] = sum / (1e-6f + cnt);
    }
}

extern "C" void kernel_launch(void* const* d_in, const int* in_sizes, int n_in,
                              void* d_out, int out_size, void* d_ws, size_t ws_size,
                              hipStream_t stream) {
    const float* feats   = (const float*)d_in[0];  // (2,4,64,64,64)
    const float* intrins = (const float*)d_in[1];  // (2,4,4,4)
    const float* extrins = (const float*)d_in[2];  // (2,4,4,4)
    float* out = (float*)d_out;                    // (2, 512, 128, 128)
    float* M   = (float*)d_ws;                     // 8 cams x 12 floats = 384 B

    bev_setup_kernel<<<1, 8, 0, stream>>>(intrins, extrins, M);
    dim3 grid(NVX, NVZ, 2);
    bev_main_kernel<<<grid, 128, 0, stream>>>(feats, M, out);
}